// Reg_Loss_60095182405996
// MI455X (gfx1250) — compile-verified
//
#include <hip/hip_runtime.h>

typedef float v2f __attribute__((ext_vector_type(2)));
typedef float v4f __attribute__((ext_vector_type(4)));
typedef float v8f __attribute__((ext_vector_type(8)));

#define DIMS   2048
#define PAIRS  (DIMS / 2)        // 1024
#define TPB    512               // 16 wave32s; covers one full row (512 float4)
#define BLOCKS 256               // 64 contiguous rows per block

// Stage 1: stream x,y with non-temporal b128 loads; per-thread register
// accumulation of 2 pair-cosine sums over the block's row chunk; combine
// across blocks with fp32 L2 atomics (256 adds per address).
__global__ __launch_bounds__(TPB) void reg_pair_sum(
    const float* __restrict__ x, const float* __restrict__ y,
    float* __restrict__ pair_acc, int rows_per_block, int batch)
{
    const int t    = threadIdx.x;                    // float4 column 0..511
    const int row0 = blockIdx.x * rows_per_block;
    int row_end    = row0 + rows_per_block;
    if (row_end > batch) row_end = batch;

    const float* px = x + (size_t)row0 * DIMS + (size_t)t * 4;
    const float* py = y + (size_t)row0 * DIMS + (size_t)t * 4;

    float acc0 = 0.0f, acc1 = 0.0f;

    #pragma unroll 4
    for (int row = row0; row < row_end; ++row) {
        v4f xv = __builtin_nontemporal_load((const v4f*)px);
        v4f yv = __builtin_nontemporal_load((const v4f*)py);
        px += DIMS;
        py += DIMS;

        // pair 0 = elements (0,1) of the float4
        float d0  = xv.x * yv.x + xv.y * yv.y;
        float nx0 = xv.x * xv.x + xv.y * xv.y;
        float ny0 = yv.x * yv.x + yv.y * yv.y;
        acc0 += d0 * __builtin_amdgcn_rsqf(nx0 * ny0);   // v_rsq_f32 (TRANS, co-issues)

        // pair 1 = elements (2,3)
        float d1  = xv.z * yv.z + xv.w * yv.w;
        float nx1 = xv.z * xv.z + xv.w * xv.w;
        float ny1 = yv.z * yv.z + yv.w * yv.w;
        acc1 += d1 * __builtin_amdgcn_rsqf(nx1 * ny1);
    }

    // each thread uniquely owns pairs 2t, 2t+1 within its block
    __hip_atomic_fetch_add(&pair_acc[2 * t],     acc0,
                           __ATOMIC_RELAXED, __HIP_MEMORY_SCOPE_AGENT);
    __hip_atomic_fetch_add(&pair_acc[2 * t + 1], acc1,
                           __ATOMIC_RELAXED, __HIP_MEMORY_SCOPE_AGENT);
}

// Stage 2: single wave32. Each lane folds 32 of the 1024 pair sums into
// u = sum (s-1)^2, then one V_WMMA_F32_16X16X4_F32 with B = ones reduces
// the 32 lane partials: A[m][0]=u[m], A[m][2]=u[m+16], other K columns 0
//   -> D[m][n] = u[m] + u[m+16] for all n.
// Lane L (0..15) holds D[0..7][L] in its 8 D VGPRs; lane L (16..31) holds
// D[8..15][L-16]. Summing a lane's 8 VGPRs and adding across the two lane
// halves (shfl_xor 16) yields the full 32-partial total in every lane.
__global__ __launch_bounds__(32) void reg_finalize(
    const float* __restrict__ pair_acc, float* __restrict__ out, float scale)
{
    const int lane = threadIdx.x;   // 0..31, EXEC all-ones (no divergence)

    float u = 0.0f;
    #pragma unroll
    for (int i = 0; i < PAIRS / 32; ++i) {
        float s = pair_acc[i * 32 + lane];
        float d = s - 1.0f;
        u = __builtin_fmaf(d, d, u);
    }

    v2f a;
    a[0] = u;     // K=0 (lanes 0-15) / K=2 (lanes 16-31)
    a[1] = 0.0f;  // K=1 / K=3 column zeroed
    v2f b;
    b[0] = 1.0f;  // B = 4x16 all-ones
    b[1] = 1.0f;
    v8f c = {};   // C = 0

    v8f dm = __builtin_amdgcn_wmma_f32_16x16x4_f32(
        /*neg_a=*/false, a, /*neg_b=*/false, b,
        /*c_mod=*/(short)0, c, /*reuse_a=*/false, /*reuse_b=*/false);

    float partial = 0.0f;
    #pragma unroll
    for (int i = 0; i < 8; ++i) partial += dm[i];

    float total = partial + __shfl_xor(partial, 16, 32);

    if (lane == 0) out[0] = total * scale;
}

extern "C" void kernel_launch(void* const* d_in, const int* in_sizes, int n_in,
                              void* d_out, int out_size, void* d_ws, size_t ws_size,
                              hipStream_t stream)
{
    (void)in_sizes; (void)n_in; (void)out_size; (void)ws_size;

    const float* x  = (const float*)d_in[0];
    const float* y  = (const float*)d_in[1];
    float* out      = (float*)d_out;
    float* pair_acc = (float*)d_ws;          // 1024 fp32 accumulators

    const int batch = 16384;                 // reference shape (16384, 1, 2048)

    // zero the per-pair accumulators every call (graph-capture safe)
    hipMemsetAsync(pair_acc, 0, PAIRS * sizeof(float), stream);

    const int rows_per_block = (batch + BLOCKS - 1) / BLOCKS;   // 64
    reg_pair_sum<<<BLOCKS, TPB, 0, stream>>>(x, y, pair_acc, rows_per_block, batch);

    const float scale = 1.0f / ((float)batch * (float)PAIRS);   // /batch /npairs
    reg_finalize<<<1, 32, 0, stream>>>(pair_acc, out, scale);
}